// CausalAttentionSortNet_87033217286669
// MI455X (gfx1250) — compile-verified
//
#include <hip/hip_runtime.h>
#include <hip/hip_bf16.h>

// Problem constants
#define BATCH     4
#define HEADS     8
#define BH        32        // BATCH*HEADS
#define BUCKETS   64
#define DIM       64
#define DIMS      8         // DIM_SORT
#define SEQ       8192
#define BSZ       128       // tokens per bucket = SEQ/BUCKETS
#define KDIM      192       // 3*DIM

// LDS padding strides (floats) - chosen even (8B-aligned v2f) and bank-spread
#define BR_STR    130       // 130 % 64 = 2  -> 16 consecutive rows conflict-free
#define WT_STR    198       // 198 % 64 = 6  -> 16 rows conflict-free
#define SQ_STR    10        // 10  % 64 = 10 -> 16 rows conflict-free

typedef __attribute__((ext_vector_type(2))) float v2f;
typedef __attribute__((ext_vector_type(8))) float v8f;

// ---------------------------------------------------------------------------
// Kernel 1: per-(bh, bucket) sum over 128 tokens x 64 dims of k.
// 2048 workgroups x 256 threads, float4 (b128) streaming loads. Bandwidth-bound.
// ---------------------------------------------------------------------------
__global__ __launch_bounds__(256)
void bucket_sum_kernel(const float* __restrict__ k, float* __restrict__ bsum) {
    const int j  = blockIdx.x;   // bucket
    const int bh = blockIdx.y;   // batch*head

    const float4* kp = (const float4*)(k + ((size_t)bh * SEQ + (size_t)j * BSZ) * DIM);
    const int v  = threadIdx.x & 15;   // float4 slot within a 64-dim row (16 per row)
    const int t0 = threadIdx.x >> 4;   // token group 0..15

    float4 acc = make_float4(0.f, 0.f, 0.f, 0.f);
    #pragma unroll
    for (int t = 0; t < BSZ / 16; ++t) {
        float4 x = kp[(size_t)(t0 + t * 16) * 16 + v];
        acc.x += x.x; acc.y += x.y; acc.z += x.z; acc.w += x.w;
    }

    __shared__ float4 red[256];
    red[threadIdx.x] = acc;
    __syncthreads();
    #pragma unroll
    for (int s = 128; s >= 16; s >>= 1) {
        if ((int)threadIdx.x < s) {
            float4 a = red[threadIdx.x];
            float4 b = red[threadIdx.x + s];
            a.x += b.x; a.y += b.y; a.z += b.z; a.w += b.w;
            red[threadIdx.x] = a;
        }
        __syncthreads();
    }
    if (threadIdx.x < 16) {
        float4* bp = (float4*)(bsum + ((size_t)bh * BUCKETS + j) * DIM);
        bp[v] = red[v];
    }
}

// ---------------------------------------------------------------------------
// Kernel 2: per-bh: bucket prefix -> b_r, WMMA fp32 GEMMs (sq, sk, R),
// masked softmax + strict-lower-tri zeroing. 32 workgroups x 128 threads.
// ---------------------------------------------------------------------------
__global__ __launch_bounds__(128)
void sortnet_kernel(const float* __restrict__ k,
                    const float* __restrict__ posq,   // (HEADS, BUCKETS, DIM)
                    const float* __restrict__ posk,
                    const float* __restrict__ wsq,    // (HEADS, KDIM, DIMS)
                    const float* __restrict__ wsk,
                    const float* __restrict__ bsum,   // (BH, BUCKETS, DIM)
                    float* __restrict__ out)          // (BH, BUCKETS, BUCKETS+1)
{
    const int bh  = blockIdx.x;
    const int h   = bh % HEADS;
    const int tid = threadIdx.x;

    __shared__ float br[BUCKETS][BR_STR];       // [j][0:64]=cumavg [j][64:128]=k bnd
    __shared__ float sqm[BUCKETS][SQ_STR];
    __shared__ float skm[BUCKETS][SQ_STR];
    // W staging (phase B) and R matrix (phases C/D) are never live together.
    __shared__ union {
        struct { float wqT[16][WT_STR]; float wkT[16][WT_STR]; } w;  // 25344 B
        float Rl[BUCKETS][BUCKETS + 2];                              // 16896 B
    } u;

    const float* wq = wsq + (size_t)h * KDIM * DIMS;
    const float* wk = wsk + (size_t)h * KDIM * DIMS;
    const float* pq = posq + (size_t)h * BUCKETS * DIM;
    const float* pk = posk + (size_t)h * BUCKETS * DIM;

    // ---- Stage W^T into LDS, zero-padded to 16 columns (all 128 threads) --
    for (int i = tid; i < KDIM * 16; i += 128) {
        const int e = i >> 4;        // 0..191  (K index)
        const int f = i & 15;        // 0..15   (padded N index)
        const bool nv = (f < DIMS);
        u.w.wqT[f][e] = nv ? wq[(size_t)e * DIMS + f] : 0.f;
        u.w.wkT[f][e] = nv ? wk[(size_t)e * DIMS + f] : 0.f;
    }

    // ---- Phase A: prefix over bucket sums -> b_r rows in LDS --------------
    if (tid < DIM) {
        const int d = tid;
        float run = 0.f;
        const float* kb = k + (size_t)bh * SEQ * DIM + d;
        const float* bs = bsum + (size_t)bh * BUCKETS * DIM + d;
        for (int j = 0; j < BUCKETS; ++j) {
            float kv = kb[(size_t)j * BSZ * DIM];       // k at bucket boundary
            float S  = run + kv;                        // inclusive prefix sum
            br[j][d]       = S / (float)(j * BSZ + 1);  // cumavg sample
            br[j][DIM + d] = kv;
            run += bs[(size_t)j * DIM];
        }
    }
    __syncthreads();

    // ---- Phase B: sq = b_qi @ Wsq, sk = b_ki @ Wsk via wmma f32 16x16x4 ---
    const int lane = tid & 31;
    const int wave = tid >> 5;      // 4 waves, each owns a 16-row M tile
    const int m0   = wave * 16;
    const int Lm   = lane & 15;     // row (A) / column (B) within tile
    const int hi   = lane >> 4;     // K sub-pair select: 0 -> K{0,1}, 1 -> K{2,3}

    v8f accq = {};
    v8f acck = {};
    // kk in [0,32): A rows come from b_r (shared between sq and sk)
    #pragma unroll 4
    for (int kk = 0; kk < 32; ++kk) {
        const int e0 = kk * 4 + hi * 2;
        v2f a  = *(const v2f*)&br[m0 + Lm][e0];
        v2f bQ = *(const v2f*)&u.w.wqT[Lm][e0];
        v2f bK = *(const v2f*)&u.w.wkT[Lm][e0];
        accq = __builtin_amdgcn_wmma_f32_16x16x4_f32(false, a, false, bQ,
                                                     (short)0, accq, false, false);
        acck = __builtin_amdgcn_wmma_f32_16x16x4_f32(false, a, false, bK,
                                                     (short)0, acck, false, false);
    }
    // kk in [32,48): A rows come from the positional embeddings (global, b64)
    #pragma unroll 4
    for (int kk = 32; kk < 48; ++kk) {
        const int e0 = kk * 4 + hi * 2;
        const size_t pidx = (size_t)(m0 + Lm) * DIM + (size_t)(e0 - 2 * DIM);
        v2f aQ = *(const v2f*)&pq[pidx];
        v2f aK = *(const v2f*)&pk[pidx];
        v2f bQ = *(const v2f*)&u.w.wqT[Lm][e0];
        v2f bK = *(const v2f*)&u.w.wkT[Lm][e0];
        accq = __builtin_amdgcn_wmma_f32_16x16x4_f32(false, aQ, false, bQ,
                                                     (short)0, accq, false, false);
        acck = __builtin_amdgcn_wmma_f32_16x16x4_f32(false, aK, false, bK,
                                                     (short)0, acck, false, false);
    }
    // C/D layout: vgpr i, lanes0-15 -> M=m0+i ; lanes16-31 -> M=m0+i+8 ; N=Lm
    #pragma unroll
    for (int i = 0; i < 8; ++i) {
        const int M = m0 + i + hi * 8;
        if (Lm < DIMS) {
            sqm[M][Lm] = accq[i];
            skm[M][Lm] = acck[i];
        }
    }
    __syncthreads();

    // ---- Phase C: R[:, 1:65] = sq @ sk^T  (64x64, K=8) --------------------
    #pragma unroll
    for (int nt = 0; nt < 4; ++nt) {
        const int n0 = nt * 16;
        v8f accr = {};
        #pragma unroll
        for (int s = 0; s < 2; ++s) {
            const int k0 = s * 4 + hi * 2;
            v2f a = *(const v2f*)&sqm[m0 + Lm][k0];
            v2f b = *(const v2f*)&skm[n0 + Lm][k0];   // B[K][N] = sk[N][K]
            accr = __builtin_amdgcn_wmma_f32_16x16x4_f32(false, a, false, b,
                                                         (short)0, accr, false, false);
        }
        #pragma unroll
        for (int i = 0; i < 8; ++i) {
            const int M = m0 + i + hi * 8;
            u.Rl[M][n0 + Lm + 1] = accr[i];  // shifted: column j+1 <- sk row j
        }
    }
    __syncthreads();

    // ---- Phase D: masked softmax + tril(diag=-1) zeroing ------------------
    if (tid < BUCKETS) {
        const int row = tid;
        u.Rl[row][0] = 0.f;                     // padded sk row -> dot = 0
        float mx = -3.402823466e38f;
        for (int c = 0; c <= row; ++c) mx = fmaxf(mx, u.Rl[row][c]);
        float sum = 0.f;
        for (int c = 0; c <= row; ++c) {
            float e = __expf(u.Rl[row][c] - mx);
            u.Rl[row][c] = e;
            sum += e;
        }
        const float inv = 1.f / sum;
        float* o = out + ((size_t)bh * BUCKETS + row) * (BUCKETS + 1);
        for (int c = 0; c < BUCKETS + 1; ++c) {
            o[c] = (c < row) ? u.Rl[row][c] * inv : 0.f;
        }
    }
}

// ---------------------------------------------------------------------------
extern "C" void kernel_launch(void* const* d_in, const int* in_sizes, int n_in,
                              void* d_out, int out_size, void* d_ws, size_t ws_size,
                              hipStream_t stream) {
    // inputs: 0=q (unused), 1=k, 2=q_pos_emb, 3=k_pos_emb, 4=Wsq, 5=Wsk
    const float* k   = (const float*)d_in[1];
    const float* pq  = (const float*)d_in[2];
    const float* pk  = (const float*)d_in[3];
    const float* wsq = (const float*)d_in[4];
    const float* wsk = (const float*)d_in[5];
    float* out  = (float*)d_out;
    float* bsum = (float*)d_ws;                 // BH*BUCKETS*DIM floats = 512 KB

    bucket_sum_kernel<<<dim3(BUCKETS, BH), 256, 0, stream>>>(k, bsum);
    sortnet_kernel<<<BH, 128, 0, stream>>>(k, pq, pk, wsq, wsk, bsum, out);
}